// ScaledDotProductAttention_24172075941944
// MI455X (gfx1250) — compile-verified
//
#include <hip/hip_runtime.h>
#include <hip/hip_bf16.h>
#include <stdint.h>

typedef __attribute__((ext_vector_type(2))) float v2f;
typedef __attribute__((ext_vector_type(4))) float v4f;
typedef __attribute__((ext_vector_type(8))) float v8f;

#define B_ 16
#define L_ 2048
#define D_ 64
#define NWAVES 8
#define COLS_PER_WAVE (L_ / NWAVES)            // 256
#define TILES_PER_WAVE (COLS_PER_WAVE / 16)    // 16

// One workgroup = one 16-row q tile of one batch. 8 waves split the 2048
// k-columns. Masked, un-normalized scores are staged in a 16x2048 LDS panel
// (128 KB of the WGP's 320 KB); O = S*V and row sums accumulate on the fly;
// a final pass scales by 1/max(rowsum,1e-14) and streams attn out with
// non-temporal 128-bit stores.
__global__ __launch_bounds__(256)
void sdpa_dot_sumnorm_kernel(const float* __restrict__ q,
                             const float* __restrict__ k,
                             const float* __restrict__ v,
                             const unsigned char* __restrict__ mask,
                             float* __restrict__ out,
                             float* __restrict__ attn)
{
    __shared__ float s_S[16 * L_];      // 128 KB masked-score panel
    __shared__ float s_O[16 * D_];      // 4 KB cross-wave O reduction
    __shared__ float s_den[16];         // row denominators

    const int tid  = threadIdx.x;
    const int wave = tid >> 5;
    const int lane = tid & 31;
    const int lh   = lane >> 4;         // which 16-lane half
    const int lm   = lane & 15;

    const int wg = blockIdx.x;
    const int b  = wg >> 7;             // / (L_/16)
    const int q0 = (wg & 127) << 4;

    const float* qb = q + ((size_t)b * L_ + q0) * D_;
    const float* kb = k + (size_t)b * L_ * D_;
    const float* vb = v + (size_t)b * L_ * D_;
    const unsigned char* mb = mask + ((size_t)b * L_ + q0) * L_;
    float* ob = out  + ((size_t)b * L_ + q0) * D_;
    float* ab = attn + ((size_t)b * L_ + q0) * L_;

    if (tid < 16) s_den[tid] = 0.0f;
    for (int i = tid; i < 16 * D_; i += 256) s_O[i] = 0.0f;
    __syncthreads();

    // Q A-fragments: 16 K-slices of 4 (A 16x4 layout: m=lm, K={2*lh,2*lh+1})
    v2f aq[16];
#pragma unroll
    for (int s = 0; s < 16; ++s)
        aq[s] = *(const v2f*)(qb + (size_t)lm * D_ + 4 * s + 2 * lh);

    v8f oacc[4];
#pragma unroll
    for (int j = 0; j < 4; ++j) oacc[j] = (v8f){0,0,0,0,0,0,0,0};
    v8f rs = (v8f){0,0,0,0,0,0,0,0};

    const int nbase = wave * COLS_PER_WAVE;

    for (int t = 0; t < TILES_PER_WAVE; ++t) {
        const int n0 = nbase + t * 16;

        // ---- scores: Q(16x64) x K^T(64x16), 16 chained f32 WMMAs ----
        v8f c = (v8f){0,0,0,0,0,0,0,0};
#pragma unroll
        for (int s = 0; s < 16; ++s) {
            // B 4x16 layout: n=lm, K rows {2*lh, 2*lh+1}; B[d][n] = K[n][d]
            v2f bf = *(const v2f*)(kb + (size_t)(n0 + lm) * D_ + 4 * s + 2 * lh);
            c = __builtin_amdgcn_wmma_f32_16x16x4_f32(false, aq[s], false, bf,
                                                      (short)0, c, false, false);
        }

        // ---- mask (NT stream), row-sum partials, stage tile in LDS ----
#pragma unroll
        for (int r = 0; r < 8; ++r) {
            const int m = r + 8 * lh;   // C/D layout: rows r / r+8 per half
            unsigned char mv =
                __builtin_nontemporal_load(mb + (size_t)m * L_ + n0 + lm);
            float sv = mv ? 0.0f : c[r];
            rs[r] += sv;
            s_S[m * L_ + n0 + lm] = sv;
        }

        // ---- O += S(16x16) x V(16x64); A re-read from LDS in A-layout ----
        v2f as[4];
#pragma unroll
        for (int kk = 0; kk < 4; ++kk)
            as[kk] = *(const v2f*)&s_S[lm * L_ + n0 + 4 * kk + 2 * lh];

#pragma unroll
        for (int j = 0; j < 4; ++j) {
#pragma unroll
            for (int kk = 0; kk < 4; ++kk) {
                const float* vp =
                    vb + (size_t)(n0 + 4 * kk + 2 * lh) * D_ + 16 * j + lm;
                v2f bf; bf.x = vp[0]; bf.y = vp[D_];
                oacc[j] = __builtin_amdgcn_wmma_f32_16x16x4_f32(
                    false, as[kk], false, bf, (short)0, oacc[j], false, false);
            }
        }
    }

    // Row-sum butterfly within each 16-lane half (masks <16 keep halves apart)
#pragma unroll
    for (int off = 1; off < 16; off <<= 1) {
#pragma unroll
        for (int r = 0; r < 8; ++r)
            rs[r] += __shfl_xor(rs[r], off, 32);
    }
    if (lm == 0) {
#pragma unroll
        for (int r = 0; r < 8; ++r)
            atomicAdd(&s_den[r + 8 * lh], rs[r]);   // ds_add_f32
    }

    // Cross-wave O reduction
#pragma unroll
    for (int j = 0; j < 4; ++j)
#pragma unroll
        for (int r = 0; r < 8; ++r)
            atomicAdd(&s_O[(r + 8 * lh) * D_ + 16 * j + lm], oacc[j][r]);

    __syncthreads();
    if (tid < 16) s_den[tid] = 1.0f / fmaxf(s_den[tid], 1e-14f);
    __syncthreads();

    // Stream normalized attn: 268 MB total across grid -> NT b128 stores
    for (int i = tid; i < 16 * (L_ / 4); i += 256) {
        const int m  = i >> 9;              // / 512 float4s per row
        const int n4 = (i & 511) << 2;
        v4f sv = *(const v4f*)&s_S[m * L_ + n4];
        sv = sv * s_den[m];
        __builtin_nontemporal_store(sv, (v4f*)(ab + (size_t)m * L_ + n4));
    }

    // Normalized output (never re-read -> NT)
    for (int i = tid; i < 16 * D_; i += 256) {
        const int m = i >> 6;
        float val = s_O[i] * s_den[m];
        __builtin_nontemporal_store(val, ob + (size_t)m * D_ + (i & 63));
    }
}

extern "C" void kernel_launch(void* const* d_in, const int* in_sizes, int n_in,
                              void* d_out, int out_size, void* d_ws, size_t ws_size,
                              hipStream_t stream) {
    const float* q = (const float*)d_in[0];
    const float* k = (const float*)d_in[1];
    const float* v = (const float*)d_in[2];
    const unsigned char* mask = (const unsigned char*)d_in[3]; // jnp.bool_ = 1 byte
    float* out  = (float*)d_out;
    float* attn = out + (size_t)B_ * L_ * D_;   // outputs concatenated in return order

    dim3 grid(B_ * (L_ / 16));   // 2048 workgroups
    dim3 block(256);             // 8 wave32
    hipLaunchKernelGGL(sdpa_dot_sumnorm_kernel, grid, block, 0, stream,
                       q, k, v, mask, out, attn);
}